// SpectralSynthesizer_23081154249466
// MI455X (gfx1250) — compile-verified
//
#include <hip/hip_runtime.h>

typedef __attribute__((ext_vector_type(2))) float v2f;
typedef __attribute__((ext_vector_type(8))) float v8f;

#define N_ROWS 8192
#define N_COLS 4096

// c[d]: tap coefficient for distance d along axis 0 (natural logs of primes, c[0]=1).
__device__ __forceinline__ float ctab(int d) {
    const float c[20] = {
        1.0f,            // d=0 (identity term: out = x + mod)
        0.0f,
        0.6931471805599453f,   // ln 2
        1.0986122886681098f,   // ln 3
        0.0f,
        1.6094379124341003f,   // ln 5
        0.0f,
        1.9459101090932196f,   // ln 7
        0.0f, 0.0f, 0.0f,
        2.3978952727983707f,   // ln 11
        0.0f,
        2.5649493574615367f,   // ln 13
        0.0f, 0.0f, 0.0f,
        2.8332133440562162f,   // ln 17
        0.0f,
        2.9444389791664403f    // ln 19
    };
    return (d >= 0 && d < 20) ? c[d] : 0.0f;
}

// out[16r+n][j] = sum_{d=0..19} c[d] * x[(16r+n-d) & 8191][j]
// One wave = one 16-row output tile x 32 column tiles, 9 V_WMMA_F32_16X16X4_F32 each.
//   M = column offset (0..15), N = output row offset (0..15), K = input row chunk.
// A 16x4 layout: lane -> M = lane%16, VGPR v -> K = v + 2*(lane/16).
// B 4x16 layout: lane -> N = lane%16, VGPR v -> K = v + 2*(lane/16).
// D 16x16 layout: lane -> N = lane%16, VGPR v -> M = v + 8*(lane/16).
__global__ __launch_bounds__(256, 2)
void spectral_synth_wmma(const float* __restrict__ x, float* __restrict__ out) {
    const int lane = threadIdx.x & 31;
    const int wv   = threadIdx.x >> 5;   // wave in block: 0..7
    const int n    = lane & 15;
    const int h    = lane >> 4;
    const int r    = blockIdx.x;         // row tile: 0..511

    // Per-wave B coefficients: B[K][n] = c[n + 19 - 4t - K].
    v2f bco[9];
#pragma unroll
    for (int t = 0; t < 9; ++t) {
        const int d0 = n + 19 - 4 * t - 2 * h;
        bco[t].x = ctab(d0);
        bco[t].y = ctab(d0 - 1);
    }

    float* const obase = out + (size_t)(r * 16 + n) * N_COLS + 8 * h + wv * 16;

    if (r >= 2 && r <= 510) {
        // Interior fast path: rows r*16-19 .. r*16+16 never wrap, so all 18
        // loads are immediate offsets from ONE per-lane base address
        // (byte imm max 35*4096*4 = 573440 < 2^23, fits signed 24-bit IOFFSET).
        const int rowbase = (r * 16 - 19 + 2 * h) * N_COLS;
#pragma unroll 1
        for (int s = 0; s < 32; ++s) {
            const int col = (s * 8 + wv) * 16 + n;
            const float* __restrict__ p = x + rowbase + col;
            __builtin_prefetch(p + 128, 0, 1);  // next column tile of this wave

            v8f acc = {};
#pragma unroll
            for (int t = 0; t < 9; ++t) {
                v2f a;
                a.x = p[t * (4 * N_COLS)];            // K = 0 + 2h
                a.y = p[t * (4 * N_COLS) + N_COLS];   // K = 1 + 2h
                acc = __builtin_amdgcn_wmma_f32_16x16x4_f32(
                    false, a, false, bco[t], (short)0, acc, false, false);
            }
            *(v8f*)(obase + s * 128) = acc;  // 32B-aligned -> 2x global_store_b128
        }
    } else {
        // Boundary path (r = 0, 1, 511): generic mod-8192 row offsets.
        int rowoff[18];
#pragma unroll
        for (int t = 0; t < 9; ++t) {
#pragma unroll
            for (int k = 0; k < 2; ++k) {
                const int row = (r * 16 - 19 + 4 * t + k + 2 * h + N_ROWS) & (N_ROWS - 1);
                rowoff[2 * t + k] = row * N_COLS;
            }
        }
#pragma unroll 1
        for (int s = 0; s < 32; ++s) {
            const int col = (s * 8 + wv) * 16 + n;
            v8f acc = {};
#pragma unroll
            for (int t = 0; t < 9; ++t) {
                v2f a;
                a.x = x[rowoff[2 * t]     + col];
                a.y = x[rowoff[2 * t + 1] + col];
                acc = __builtin_amdgcn_wmma_f32_16x16x4_f32(
                    false, a, false, bco[t], (short)0, acc, false, false);
            }
            *(v8f*)(obase + s * 128) = acc;
        }
    }
}

extern "C" void kernel_launch(void* const* d_in, const int* in_sizes, int n_in,
                              void* d_out, int out_size, void* d_ws, size_t ws_size,
                              hipStream_t stream) {
    (void)in_sizes; (void)n_in; (void)d_ws; (void)ws_size; (void)out_size;
    const float* x = (const float*)d_in[0];
    float* out = (float*)d_out;
    dim3 grid(N_ROWS / 16);  // 512 row tiles
    dim3 block(256);         // 8 waves; each wave -> 1 row tile x 32 col tiles
    spectral_synth_wmma<<<grid, block, 0, stream>>>(x, out);
}